// CoConv_90391881711982
// MI455X (gfx1250) — compile-verified
//
#include <hip/hip_runtime.h>
#include <hip/hip_bf16.h>
#include <stdint.h>

#define D_FEAT 128
#define WAVES_PER_BLOCK 8
#define BLOCK_THREADS (WAVES_PER_BLOCK * 32)
#define EPW 4 // edges gathered per wave per chunk (must match s_wait_asynccnt literal)

__global__ __launch_bounds__(BLOCK_THREADS)
void coconv_edge_scatter_kernel(const float* __restrict__ emb,
                                const int* __restrict__ src,
                                const int* __restrict__ dst,
                                float* __restrict__ out,
                                unsigned int* __restrict__ deg,
                                int E) {
    // Double-buffered per-wave staging: 2 bufs x 8 waves x 4 rows x 512B = 32 KB.
    __shared__ float tile[2][WAVES_PER_BLOCK][EPW][D_FEAT];

    const int lane = threadIdx.x & 31;
    const int wave = threadIdx.x >> 5;
    // All counts fit in int32: E = 600k, nchunks = 150k.
    const int gwave   = blockIdx.x * WAVES_PER_BLOCK + wave;
    const int nwaves  = gridDim.x * WAVES_PER_BLOCK;
    const int nchunks = (E + EPW - 1) / EPW;
    if (gwave >= nchunks) return;

    const unsigned long long embBase = (unsigned long long)(uintptr_t)emb;

    // Issue EPW async row-gathers into LDS buffer `buf`. Always issues exactly
    // EPW instructions (tail edges clamped to a dummy) so ASYNCcnt bookkeeping
    // stays exact for the double-buffer waits.
    auto issue = [&](int chunk, int buf) {
        int base = chunk * EPW;
#pragma unroll
        for (int k = 0; k < EPW; ++k) {
            int e = base + k;
            if (e >= E) e = E - 1; // dummy gather, never consumed
            int s = src[e];        // wave-uniform value
            unsigned long long ga =
                embBase + ((unsigned long long)(unsigned int)s << 9) + // 512B rows
                (unsigned long long)(lane * 16);
            unsigned int lo =
                (unsigned int)(uintptr_t)(&tile[buf][wave][k][0]) +
                (unsigned int)(lane * 16);
            // 32 lanes x 16B = one full 512B feature row per instruction.
            asm volatile("global_load_async_to_lds_b128 %0, %1, off"
                         :: "v"(lo), "v"(ga)
                         : "memory");
        }
    };

    // Scatter-add buffer `buf` rows into out[] with coalesced f32 atomics:
    // instruction q touches the contiguous 128B line [32*q, 32*q+32) of the row.
    auto consume = [&](int chunk, int buf) {
        int base = chunk * EPW;
#pragma unroll
        for (int k = 0; k < EPW; ++k) {
            int e = base + k;
            if (e < E) {
                int dn = dst[e]; // wave-uniform value
                float* orow = out + ((size_t)(unsigned int)dn << 7); // 128 floats
                const float* lrow = &tile[buf][wave][k][0];
#pragma unroll
                for (int q = 0; q < 4; ++q) {
                    float v = lrow[lane + 32 * q];
                    (void)__hip_atomic_fetch_add(orow + lane + 32 * q, v,
                                                 __ATOMIC_RELAXED,
                                                 __HIP_MEMORY_SCOPE_AGENT);
                }
                if (lane == 0) {
                    (void)__hip_atomic_fetch_add(deg + dn, 1u,
                                                 __ATOMIC_RELAXED,
                                                 __HIP_MEMORY_SCOPE_AGENT);
                }
            }
        }
    };

    int c = gwave;
    int buf = 0;
    issue(c, buf);
    for (;;) {
        int cn = c + nwaves;
        if (cn < nchunks) {
            issue(cn, buf ^ 1);
            // Async loads complete in order per wave: <=EPW outstanding means
            // the previous chunk's EPW loads (current buffer) have landed.
            asm volatile("s_wait_asynccnt 4" ::: "memory");
            consume(c, buf);
            c = cn;
            buf ^= 1;
        } else {
            asm volatile("s_wait_asynccnt 0" ::: "memory");
            consume(c, buf);
            break;
        }
    }
}

__global__ void coconv_zero_kernel(float* __restrict__ out,
                                   unsigned int* __restrict__ deg,
                                   int nOut, int nDeg) {
    int i = blockIdx.x * blockDim.x + threadIdx.x;
    if (i < nOut) out[i] = 0.0f;
    if (i < nDeg) deg[i] = 0u;
}

__global__ void coconv_finalize_kernel(float* __restrict__ out,
                                       const unsigned int* __restrict__ deg,
                                       int nOut) {
    int i = blockIdx.x * blockDim.x + threadIdx.x;
    if (i < nOut) {
        float d = (float)deg[i >> 7]; // D_FEAT == 128
        out[i] = out[i] / fmaxf(d, 1.0f);
    }
}

extern "C" void kernel_launch(void* const* d_in, const int* in_sizes, int n_in,
                              void* d_out, int out_size, void* d_ws, size_t ws_size,
                              hipStream_t stream) {
    const float* emb = (const float*)d_in[0];
    const int* src   = (const int*)d_in[1];
    const int* dst   = (const int*)d_in[2];
    (void)n_in; (void)out_size; (void)ws_size;

    const int E = in_sizes[1];
    const int N = in_sizes[0] / D_FEAT;
    float* out = (float*)d_out;
    unsigned int* deg = (unsigned int*)d_ws; // N uint32 of scratch

    const int nOut = N * D_FEAT;
    const int zThreads = 256;
    const int zBlocks = (nOut + zThreads - 1) / zThreads;
    coconv_zero_kernel<<<zBlocks, zThreads, 0, stream>>>(out, deg, nOut, N);

    const int nchunks = (E + EPW - 1) / EPW;
    int maxBlocks = (nchunks + WAVES_PER_BLOCK - 1) / WAVES_PER_BLOCK;
    int blocks = 4096; // ~32k waves -> several chunks/wave so the pipeline fills
    if (blocks > maxBlocks) blocks = maxBlocks;
    if (blocks > 0) {
        coconv_edge_scatter_kernel<<<blocks, BLOCK_THREADS, 0, stream>>>(
            emb, src, dst, out, deg, E);
    }

    coconv_finalize_kernel<<<zBlocks, zThreads, 0, stream>>>(out, deg, nOut);
}